// AGMBrain_79826262164138
// MI455X (gfx1250) — compile-verified
//
#include <hip/hip_runtime.h>

// ---------------------------------------------------------------------------
// AGM brain fused kernel for gfx1250 (MI455X), round 2.
// TB=16 batch elements per workgroup, states resident in 320 KB LDS.
// All matmuls on V_WMMA_F32_16X16X4_F32 with full 16-row M utilization and
// multi-accumulator K-loops to keep loads and WMMAs overlapped.
// ---------------------------------------------------------------------------

typedef float v2f __attribute__((ext_vector_type(2)));
typedef float v8f __attribute__((ext_vector_type(8)));

#define BATCH    16384
#define DI       256     // INPUT_DIM
#define NN       32      // N_NEURONS
#define DD       128     // NEURON_DIM
#define NSTEPS   3
#define TB       16      // batch tile per workgroup (= WMMA M)
#define NWAVES   16
#define NTHREADS (NWAVES * 32)

#define ST_FLOATS (TB * NN * DD)   // 65536 floats = 256 KB (LDS states)
#define SC_FLOATS (TB * NN * NN)   // 16384 floats =  64 KB (LDS scores / staging)

__device__ __forceinline__ v8f wmma_f32(v2f a, v2f b, v8f c) {
  return __builtin_amdgcn_wmma_f32_16x16x4_f32(
      /*neg_a=*/false, a, /*neg_b=*/false, b,
      /*c_mod=*/(short)0, c, /*reuse_a=*/false, /*reuse_b=*/false);
}

__global__ __launch_bounds__(NTHREADS, 1)
void agm_brain_kernel(const float* __restrict__ x,            // (B, DI)
                      const float* __restrict__ W_in,         // (DI, DD)
                      const float* __restrict__ b_in,         // (DD)
                      const float* __restrict__ neuron_states,// (NN, DD)
                      const float* __restrict__ E,            // (NN, NN, DD)
                      const float* __restrict__ W_rec,        // (DD, DI)
                      const float* __restrict__ b_rec,        // (DI)
                      const float* __restrict__ W_score,      // (DD)
                      const float* __restrict__ b_score,      // (1)
                      float* __restrict__ out_rec,            // (B, DI)
                      float* __restrict__ out_score)          // (B)
{
  __shared__ float sST[ST_FLOATS];   // states [TB][NN][DD]
  __shared__ float sSC[SC_FLOATS];   // scores [TB][NN][NN] (also x_t staging)

  const int tid   = threadIdx.x;
  const int lane  = tid & 31;
  const int w     = tid >> 5;        // wave id 0..15
  const int m16   = lane & 15;       // M index (A/C) or N index (B/C)
  const int khalf = lane >> 4;       // 0 or 1
  const int koff  = khalf * 2;       // K pair offset within k0..k0+3
  const int gb    = blockIdx.x * TB; // global batch base

  // Warm L2 with the edge tensor (512 KB, shared by all workgroups).
  for (int r = tid; r < NN * NN; r += NTHREADS)
    __builtin_prefetch(E + (size_t)r * DD, 0, 3);

  // ---- Phase 0: x_t = x @ W_in + b_in  (M=TB=16, N=DD, K=DI) ----
  // Wave pairs (w, w+8) share an N-tile and split the K=256 range in half;
  // partials are combined through the staging buffer.
  {
    const int d0    = 16 * (w & 7);
    const int kbase = (w >> 3) * (DI / 2);
    v8f a0 = {}, a1 = {}, a2 = {}, a3 = {};
    #pragma unroll
    for (int ks = 0; ks < DI / 8; ks += 4) {      // 32 k-steps, 4 accumulators
      #pragma unroll
      for (int u = 0; u < 4; ++u) {
        const int k = kbase + 4 * (ks + u) + koff;
        v2f a = *reinterpret_cast<const v2f*>(x + (size_t)(gb + m16) * DI + k);
        v2f b;
        b.x = W_in[(k    ) * DD + d0 + m16];
        b.y = W_in[(k + 1) * DD + d0 + m16];
        v8f& acc = (u == 0) ? a0 : (u == 1) ? a1 : (u == 2) ? a2 : a3;
        acc = wmma_f32(a, b, acc);
      }
    }
    const v8f acc = (a0 + a1) + (a2 + a3);
    if (w < 8) {                                  // first K-half: store partial
      #pragma unroll
      for (int r = 0; r < 8; ++r)
        sSC[(khalf * 8 + r) * DD + d0 + m16] = acc[r];
    }
    __syncthreads();
    if (w >= 8) {                                 // second K-half: combine + bias
      #pragma unroll
      for (int r = 0; r < 8; ++r)
        sSC[(khalf * 8 + r) * DD + d0 + m16] += acc[r] + b_in[d0 + m16];
    }
  }
  __syncthreads();

  // ---- Phase 0b: states[b][i][d] = neuron_states[i][d] + x_t[b][d] ----
  for (int e = tid; e < ST_FLOATS; e += NTHREADS) {
    const int b   = e >> 12;         // / (NN*DD)
    const int rem = e & 4095;
    const int i   = rem >> 7;        // / DD
    const int d   = rem & 127;
    sST[e] = neuron_states[i * DD + d] + sSC[b * DD + d];
  }
  __syncthreads();

  // ---- Recurrent steps ----
  for (int step = 0; step < NSTEPS; ++step) {
    // (a) scores[b][i][j] = sum_q E[i][j][q] * states[b][i][q]; zero diagonal.
    //     Per fixed i: GEMM M=batch(16), N=j(32 -> 2 tiles), K=q(128).
    for (int t = 0; t < 4; ++t) {
      const int jidx = w * 4 + t;    // 64 (i, jtile) jobs over 16 waves
      const int i    = jidx >> 1;
      const int jt   = jidx & 1;
      const int j    = jt * 16 + m16;
      v8f a0 = {}, a1 = {}, a2 = {}, a3 = {};
      #pragma unroll
      for (int ks = 0; ks < DD / 4; ks += 4) {    // 32 k-steps, 4 accumulators
        #pragma unroll
        for (int u = 0; u < 4; ++u) {
          const int k = 4 * (ks + u) + koff;
          v2f a = *reinterpret_cast<const v2f*>(&sST[m16 * (NN * DD) + i * DD + k]);
          v2f b = *reinterpret_cast<const v2f*>(E + ((size_t)(i * NN + j)) * DD + k);
          v8f& acc = (u == 0) ? a0 : (u == 1) ? a1 : (u == 2) ? a2 : a3;
          acc = wmma_f32(a, b, acc);
        }
      }
      v8f acc = (a0 + a1) + (a2 + a3);
      if (j == i) {                  // off-diagonal mask: kill column j==i
        #pragma unroll
        for (int r = 0; r < 8; ++r) acc[r] = 0.0f;
      }
      #pragma unroll
      for (int r = 0; r < 8; ++r)
        sSC[(khalf * 8 + r) * (NN * NN) + i * NN + j] = acc[r];
    }
    __syncthreads();

    // (b) states[b] = relu(scores[b] @ states[b]); wave w owns batch b == w.
    //     M=i(32 -> 2 tiles), N=p(128 -> 8 tiles), K=j(32). All 16 output
    //     tiles held in registers until every read of old states is done,
    //     then written back in-place (wave-private -> no barrier needed).
    v8f accv[16];
    #pragma unroll
    for (int t = 0; t < 16; ++t) {
      const int it = t >> 3;
      const int pt = t & 7;
      v8f a0 = {}, a1 = {};
      #pragma unroll
      for (int ks = 0; ks < NN / 4; ks += 2) {    // 8 k-steps, 2 accumulators
        #pragma unroll
        for (int u = 0; u < 2; ++u) {
          const int k = 4 * (ks + u) + koff;
          v2f a = *reinterpret_cast<const v2f*>(
              &sSC[w * (NN * NN) + (it * 16 + m16) * NN + k]);
          v2f b;
          b.x = sST[w * (NN * DD) + (k    ) * DD + pt * 16 + m16];
          b.y = sST[w * (NN * DD) + (k + 1) * DD + pt * 16 + m16];
          v8f& acc = (u == 0) ? a0 : a1;
          acc = wmma_f32(a, b, acc);
        }
      }
      accv[t] = a0 + a1;
    }
    #pragma unroll
    for (int t = 0; t < 16; ++t) {
      const int it = t >> 3;
      const int pt = t & 7;
      #pragma unroll
      for (int r = 0; r < 8; ++r) {
        const int i = it * 16 + khalf * 8 + r;
        const int p = pt * 16 + m16;
        const float v = accv[t][r];
        sST[w * (NN * DD) + i * DD + p] = v > 0.0f ? v : 0.0f;
      }
    }
    __syncthreads();
  }

  // ---- Phase 2: out = states[:, NN-1, :]; recreation = out@W_rec + b_rec ----
  {
    const int e0 = 16 * w;           // 16 N-tiles cover DI=256, one per wave
    v8f a0 = {}, a1 = {}, a2 = {}, a3 = {};
    #pragma unroll
    for (int ks = 0; ks < DD / 4; ks += 4) {
      #pragma unroll
      for (int u = 0; u < 4; ++u) {
        const int k = 4 * (ks + u) + koff;
        v2f a = *reinterpret_cast<const v2f*>(
            &sST[m16 * (NN * DD) + (NN - 1) * DD + k]);
        v2f b;
        b.x = W_rec[(k    ) * DI + e0 + m16];
        b.y = W_rec[(k + 1) * DI + e0 + m16];
        v8f& acc = (u == 0) ? a0 : (u == 1) ? a1 : (u == 2) ? a2 : a3;
        acc = wmma_f32(a, b, acc);
      }
    }
    const v8f acc = (a0 + a1) + (a2 + a3);
    #pragma unroll
    for (int r = 0; r < 8; ++r) {
      const int m = khalf * 8 + r;
      out_rec[(size_t)(gb + m) * DI + e0 + m16] = acc[r] + b_rec[e0 + m16];
    }
  }

  // score_out = out @ W_score + b_score  (tiny: 16 dots of length 128)
  if (tid < TB) {
    float s = 0.0f;
    #pragma unroll 4
    for (int d = 0; d < DD; ++d)
      s += sST[tid * (NN * DD) + (NN - 1) * DD + d] * W_score[d];
    out_score[gb + tid] = s + b_score[0];
  }
}

extern "C" void kernel_launch(void* const* d_in, const int* in_sizes, int n_in,
                              void* d_out, int out_size, void* d_ws, size_t ws_size,
                              hipStream_t stream) {
  const float* x        = (const float*)d_in[0];
  const float* W_in     = (const float*)d_in[1];
  const float* b_in     = (const float*)d_in[2];
  const float* ns       = (const float*)d_in[3];
  const float* E        = (const float*)d_in[4];
  const float* W_rec    = (const float*)d_in[5];
  const float* b_rec    = (const float*)d_in[6];
  const float* W_score  = (const float*)d_in[7];
  const float* b_score  = (const float*)d_in[8];

  float* out_rec   = (float*)d_out;                        // (B, DI) first
  float* out_score = (float*)d_out + (size_t)BATCH * DI;   // then (B, 1)

  dim3 grid(BATCH / TB);
  dim3 block(NTHREADS);
  hipLaunchKernelGGL(agm_brain_kernel, grid, block, 0, stream,
                     x, W_in, b_in, ns, E, W_rec, b_rec, W_score, b_score,
                     out_rec, out_score);
}